// GConv_584115552914
// MI455X (gfx1250) — compile-verified
//
#include <hip/hip_runtime.h>
#include <hip/hip_bf16.h>

typedef __attribute__((ext_vector_type(2))) float v2f;
typedef __attribute__((ext_vector_type(8))) float v8f;
typedef __attribute__((ext_vector_type(4))) unsigned int v4u;
typedef __attribute__((ext_vector_type(8))) int v8i_t;
typedef __attribute__((ext_vector_type(4))) int v4i_t;

#define FD 128               // feature dim (D == H == 128)
#define LDS_STRIDE 132       // padded K-stride (4m..4m+3 banks/lane -> conflict-free b64)
#define ROWS_PB 64           // rows of X per block
// LDS: Wt (128 cols x 132) + X tile (64 rows x 132)
#define GEMM_LDS_BYTES ((FD + ROWS_PB) * LDS_STRIDE * 4)

// ---------------- degree / normalization ----------------
__global__ void deg_init_kernel(float* __restrict__ deg, int n) {
    int i = blockIdx.x * blockDim.x + threadIdx.x;
    if (i < n) deg[i] = 1.0f;              // self-loop weight
}

__global__ void deg_acc_kernel(const int* __restrict__ dst,
                               const float* __restrict__ ew,
                               float* __restrict__ deg, int e) {
    int i = blockIdx.x * blockDim.x + threadIdx.x;
    if (i < e) atomicAdd(&deg[dst[i]], ew[i]);
}

__global__ void deg_rsqrt_kernel(const float* __restrict__ deg,
                                 float* __restrict__ dinv, int n) {
    int i = blockIdx.x * blockDim.x + threadIdx.x;
    if (i < n) {
        float d = deg[i];
        dinv[i] = (d > 0.0f) ? rsqrtf(d) : 0.0f;
    }
}

__global__ void zero_kernel(float* __restrict__ p, size_t n) {
    size_t i = (size_t)blockIdx.x * blockDim.x + threadIdx.x;
    if (i < n) p[i] = 0.0f;
}

// ---------------- GEMM: h = X @ W  (fp32 WMMA 16x16x4 + TDM X-tile stage) ----
// block = 256 threads (8 waves), computes 64 rows x 128 cols.
// wave w: row-tile rt = w>>1 (16 rows), col half ct = w&1 (4 x 16 cols).
__global__ __launch_bounds__(256) void gemm_wmma_kernel(const float* __restrict__ X,
                                                        const float* __restrict__ W,
                                                        float* __restrict__ Hout,
                                                        int nrows) {
    extern __shared__ float smem[];
    float* lWt = smem;                       // W^T: lWt[n*LDS_STRIDE + k]
    float* xt  = smem + FD * LDS_STRIDE;     // X tile: xt[r*LDS_STRIDE + k]
    const int t = threadIdx.x;
    const int row0 = blockIdx.x * ROWS_PB;

#if __has_builtin(__builtin_amdgcn_tensor_load_to_lds) && __has_builtin(__builtin_amdgcn_s_wait_tensorcnt)
    // --- Tensor Data Mover: DMA 64x128 fp32 X tile -> LDS (padded to 132-float rows)
    if (t < 32) {
        unsigned long long ga = (unsigned long long)(const void*)(X + (size_t)row0 * FD);
        unsigned int ldsa = (unsigned int)(size_t)(void*)xt;   // flat LDS addr low 32b = LDS offset
        v4u g0;
        g0[0] = 1u;                                            // count=1, no gather
        g0[1] = ldsa;                                          // lds_addr (bytes)
        g0[2] = (unsigned int)ga;                              // global_addr[31:0]
        g0[3] = (unsigned int)((ga >> 32) & 0x01FFFFFFu) | (2u << 30); // ga[56:32] | type=2
        v8i_t g1;
        // data_size=4B(code 2), pad_enable, pad_interval=128 DW(code 6), pad_amount=4 DW(code 3)
        g1[0] = (int)((2u << 16) | (1u << 20) | (6u << 22) | (3u << 25));
        g1[1] = (int)(((unsigned)FD & 0xFFFFu) << 16);         // tensor_dim0[15:0]=128
        g1[2] = (int)((((unsigned)FD >> 16) & 0xFFFFu) |
                      (((unsigned)nrows & 0xFFFFu) << 16));    // tensor_dim1 lo
        g1[3] = (int)((((unsigned)nrows >> 16) & 0xFFFFu) |
                      ((unsigned)FD << 16));                   // tensor_dim1 hi | tile_dim0=128
        g1[4] = (int)((unsigned)ROWS_PB);                      // tile_dim1=64, tile_dim2=0
        g1[5] = (int)(unsigned)FD;                             // tensor_dim0_stride=128
        g1[6] = 0;                                             // stride hi | dim1_stride lo
        g1[7] = 0;
        v4i_t gz4 = {0, 0, 0, 0};
        v8i_t gz8 = {0, 0, 0, 0, 0, 0, 0, 0};
        __builtin_amdgcn_tensor_load_to_lds(g0, g1, gz4, gz4, gz8, 0);
    }
#else
    // fallback: cooperative vector load of X tile
    for (int i = 0; i < (ROWS_PB * FD / 4) / 256; ++i) {
        int idx = t + i * 256;                  // float4 index in tile
        int r = idx >> 5;
        int c = (idx & 31) << 2;
        int gr = row0 + r; if (gr >= nrows) gr = nrows - 1;
        float4 v = *(const float4*)(X + (size_t)gr * FD + c);
        xt[r * LDS_STRIDE + c + 0] = v.x;
        xt[r * LDS_STRIDE + c + 1] = v.y;
        xt[r * LDS_STRIDE + c + 2] = v.z;
        xt[r * LDS_STRIDE + c + 3] = v.w;
    }
#endif

    // --- cooperative load + transpose of W (128x128 fp32) into LDS (overlaps TDM)
    const float4* Wv = (const float4*)W;
#pragma unroll
    for (int i = 0; i < 16; ++i) {
        int idx = t + i * 256;                  // float4 index into W
        float4 v = Wv[idx];
        int k = idx >> 5;                       // W row (K)
        int n = (idx & 31) << 2;                // W col (N)
        lWt[(n + 0) * LDS_STRIDE + k] = v.x;
        lWt[(n + 1) * LDS_STRIDE + k] = v.y;
        lWt[(n + 2) * LDS_STRIDE + k] = v.z;
        lWt[(n + 3) * LDS_STRIDE + k] = v.w;
    }
#if __has_builtin(__builtin_amdgcn_tensor_load_to_lds) && __has_builtin(__builtin_amdgcn_s_wait_tensorcnt)
    if (t < 32) __builtin_amdgcn_s_wait_tensorcnt(0);
#endif
    __syncthreads();

    const int lane = t & 31;
    const int wave = t >> 5;                    // 0..7
    const int rt   = wave >> 1;                 // row tile 0..3
    const int ct   = wave & 1;                  // column half 0..1
    const int m    = lane & 15;
    const int kk   = (lane >> 4) << 1;          // 0 or 2

    const float* arow = xt + (size_t)(rt * 16 + m) * LDS_STRIDE;
    const float* b0c  = lWt + (size_t)(ct * 64 +  0 + m) * LDS_STRIDE;
    const float* b1c  = lWt + (size_t)(ct * 64 + 16 + m) * LDS_STRIDE;
    const float* b2c  = lWt + (size_t)(ct * 64 + 32 + m) * LDS_STRIDE;
    const float* b3c  = lWt + (size_t)(ct * 64 + 48 + m) * LDS_STRIDE;

    v8f c0 = {}, c1 = {}, c2 = {}, c3 = {};
#pragma unroll 8
    for (int k = 0; k < FD; k += 4) {
        v2f a  = *(const v2f*)(arow + k + kk);
        v2f b0 = *(const v2f*)(b0c + k + kk);
        v2f b1 = *(const v2f*)(b1c + k + kk);
        v2f b2 = *(const v2f*)(b2c + k + kk);
        v2f b3 = *(const v2f*)(b3c + k + kk);
        c0 = __builtin_amdgcn_wmma_f32_16x16x4_f32(false, a, false, b0, (short)0, c0, false, false);
        c1 = __builtin_amdgcn_wmma_f32_16x16x4_f32(false, a, false, b1, (short)0, c1, false, false);
        c2 = __builtin_amdgcn_wmma_f32_16x16x4_f32(false, a, false, b2, (short)0, c2, false, false);
        c3 = __builtin_amdgcn_wmma_f32_16x16x4_f32(false, a, false, b3, (short)0, c3, false, false);
    }

    // C/D layout: VGPR r -> row (r + 8*(lane>=16)), col = lane&15
    const int cm = (lane >> 4) << 3;
    const int cn = lane & 15;
#pragma unroll
    for (int r = 0; r < 8; ++r) {
        int orow = row0 + rt * 16 + cm + r;
        if (orow < nrows) {
            float* o = Hout + (size_t)orow * FD + ct * 64 + cn;
            o[0]  = c0[r];
            o[16] = c1[r];
            o[32] = c2[r];
            o[48] = c3[r];
        }
    }
}

// ---------------- edge scatter: agg[dst] += h[src] * norm ----------------
__global__ __launch_bounds__(256) void agg_edges_kernel(const float* __restrict__ Hf,
                                                        const int* __restrict__ src,
                                                        const int* __restrict__ dst,
                                                        const float* __restrict__ ew,
                                                        const float* __restrict__ dinv,
                                                        float* __restrict__ agg, int e) {
    int gwave = (int)(((size_t)blockIdx.x * blockDim.x + threadIdx.x) >> 5);
    int lane  = threadIdx.x & 31;
    if (gwave >= e) return;
    int s = src[gwave];
    int d = dst[gwave];
    float norm = dinv[s] * ew[gwave] * dinv[d];
    const float4* hs = (const float4*)(Hf + (size_t)s * FD);
    float4 v = hs[lane];
    float* o = agg + (size_t)d * FD + lane * 4;
    atomicAdd(o + 0, v.x * norm);
    atomicAdd(o + 1, v.y * norm);
    atomicAdd(o + 2, v.z * norm);
    atomicAdd(o + 3, v.w * norm);
}

// ---------------- self-loop + bias + relu ----------------
__global__ void finish_kernel(const float* __restrict__ agg,
                              const float* __restrict__ Hf,
                              const float* __restrict__ dinv,
                              const float* __restrict__ bias,
                              float* __restrict__ out, int n) {
    size_t i = (size_t)blockIdx.x * blockDim.x + threadIdx.x;
    size_t tot = (size_t)n * FD;
    if (i >= tot) return;
    int node = (int)(i >> 7);
    int f    = (int)(i & 127);
    float di = dinv[node];
    float v = agg[i] + Hf[i] * di * di + bias[f];
    out[i] = fmaxf(v, 0.0f);
}

extern "C" void kernel_launch(void* const* d_in, const int* in_sizes, int n_in,
                              void* d_out, int out_size, void* d_ws, size_t ws_size,
                              hipStream_t stream) {
    const float* x  = (const float*)d_in[0];
    const int*   ei = (const int*)  d_in[1];
    const float* ew = (const float*)d_in[2];
    const float* W1 = (const float*)d_in[3];
    const float* b1 = (const float*)d_in[4];
    const float* W2 = (const float*)d_in[5];
    const float* b2 = (const float*)d_in[6];

    const int N = in_sizes[0] / FD;
    const int E = in_sizes[2];
    const int* src = ei;
    const int* dst = ei + E;

    float* ws   = (float*)d_ws;
    size_t npad = ((size_t)N + 255) & ~(size_t)255;
    float* deg  = ws;
    float* dinv = ws + npad;
    float* h    = dinv + npad;
    float* agg  = h + (size_t)N * FD;
    float* out  = (float*)d_out;
    const size_t nh = (size_t)N * FD;

    // degree + symmetric normalization
    deg_init_kernel <<<(N + 255) / 256, 256, 0, stream>>>(deg, N);
    deg_acc_kernel  <<<(E + 255) / 256, 256, 0, stream>>>(dst, ew, deg, E);
    deg_rsqrt_kernel<<<(N + 255) / 256, 256, 0, stream>>>(deg, dinv, N);

    const int gemm_grid = (N + ROWS_PB - 1) / ROWS_PB;
    const int elem_grid = (int)((nh + 255) / 256);
    const int edge_grid = (E + 7) / 8;

    // ---- layer 1: in = x, z1 -> d_out ----
    zero_kernel     <<<elem_grid, 256, 0, stream>>>(agg, nh);
    gemm_wmma_kernel<<<gemm_grid, 256, GEMM_LDS_BYTES, stream>>>(x, W1, h, N);
    agg_edges_kernel<<<edge_grid, 256, 0, stream>>>(h, src, dst, ew, dinv, agg, E);
    finish_kernel   <<<elem_grid, 256, 0, stream>>>(agg, h, dinv, b1, out, N);

    // ---- layer 2: in = z1 (d_out), final -> d_out ----
    zero_kernel     <<<elem_grid, 256, 0, stream>>>(agg, nh);
    gemm_wmma_kernel<<<gemm_grid, 256, GEMM_LDS_BYTES, stream>>>(out, W2, h, N);
    agg_edges_kernel<<<edge_grid, 256, 0, stream>>>(h, src, dst, ew, dinv, agg, E);
    finish_kernel   <<<elem_grid, 256, 0, stream>>>(agg, h, dinv, b2, out, N);
}